// Discriminator_54511724921068
// MI455X (gfx1250) — compile-verified
//
#include <hip/hip_runtime.h>
#include <hip/hip_bf16.h>
#include <math.h>

#define B_      2
#define N_      4096
#define CIN_    16
#define KNN_    20
#define RATE_   2
#define KR_     (KNN_*RATE_)     // 40
#define NK_     (N_*KNN_)        // 81920 columns per batch
#define GROUPS_ 32
#define EPS_GN  1e-5f
#define EPS_PD  1e-6f

typedef __attribute__((ext_vector_type(2))) float v2f;
typedef __attribute__((ext_vector_type(8))) float v8f;

// ---------------------------------------------------------------------------
// kNN (dilated): one thread per (b,n), maintains sorted top-40 by insertion,
// emits every RATE-th index (ranks 0,2,...,38).
// ---------------------------------------------------------------------------
__global__ void knn_kernel(const float* __restrict__ pos, int* __restrict__ idx)
{
    int t = blockIdx.x * blockDim.x + threadIdx.x;
    if (t >= B_ * N_) return;
    int b = t / N_;
    int n = t - b * N_;
    const float* p = pos + (size_t)b * 3 * N_;
    float px = p[n], py = p[N_ + n], pz = p[2 * N_ + n];

    float dk[KR_];
    int   ik[KR_];
    #pragma unroll
    for (int i = 0; i < KR_; ++i) { dk[i] = 3.0e38f; ik[i] = 0; }

    for (int m = 0; m < N_; ++m) {
        float dx = px - p[m];
        float dy = py - p[N_ + m];
        float dz = pz - p[2 * N_ + m];
        float d2 = dx * dx + dy * dy + dz * dz;
        if (d2 < dk[KR_ - 1]) {
            int q = KR_ - 1;
            while (q > 0 && dk[q - 1] > d2) {
                dk[q] = dk[q - 1]; ik[q] = ik[q - 1]; --q;
            }
            dk[q] = d2; ik[q] = m;
        }
    }
    int* o = idx + (size_t)t * KNN_;
    #pragma unroll
    for (int kk = 0; kk < KNN_; ++kk) o[kk] = ik[kk * RATE_];
}

// ---------------------------------------------------------------------------
// graph_feature: channels = [x_j - x_n (Cin) | pos_j - pos_n (3) | dist (1) |
//                            x_n (Cin)] ; layout h[b][c][n*K + kk]
// ---------------------------------------------------------------------------
__global__ void graph_feature_kernel(const float* __restrict__ x,
                                     const float* __restrict__ pos,
                                     const int* __restrict__ idx,
                                     float* __restrict__ h, int Cin)
{
    int t = blockIdx.x * blockDim.x + threadIdx.x;
    if (t >= B_ * N_ * KNN_) return;
    int b  = t / (N_ * KNN_);
    int r  = t - b * (N_ * KNN_);
    int n  = r / KNN_;
    int kk = r - n * KNN_;
    int j  = idx[t];
    int Cfeat = 2 * Cin + 4;

    const float* xb = x   + (size_t)b * Cin * N_;
    const float* pb = pos + (size_t)b * 3 * N_;
    float* hb = h + (size_t)b * Cfeat * NK_ + (size_t)n * KNN_ + kk;

    for (int c = 0; c < Cin; ++c) {
        float xn = xb[(size_t)c * N_ + n];
        float xj = xb[(size_t)c * N_ + j];
        hb[(size_t)c * NK_]              = xj - xn;
        hb[(size_t)(Cin + 4 + c) * NK_]  = xn;
    }
    float s = 0.f;
    for (int a = 0; a < 3; ++a) {
        float pn = pb[(size_t)a * N_ + n];
        float pj = pb[(size_t)a * N_ + j];
        hb[(size_t)(Cin + a) * NK_] = pj - pn;
        float d = pn - pj + EPS_PD;
        s += d * d;
    }
    hb[(size_t)(Cin + 3) * NK_] = sqrtf(s);
}

// ---------------------------------------------------------------------------
// GEMM via V_WMMA_F32_16X16X4_F32 with 4-way M blocking:
//   one wave -> 64(M) x 16(cols) tile, 4 accumulators, B fragment loaded once
//   per K-step and reused by 4 WMMAs (vmem/wmma ~1.5, h re-read factor Cout/64).
// A 16x4: lanes 0-15 => M=0..15,K={0,1}; lanes 16-31 => K={2,3}.
// B 4x16: VGPR0 rows K={0,2}, VGPR1 rows K={1,3}; cols = lane&15.
// D 16x16: VGPR r => M=r+8*(lane>=16), N=lane&15.
// Requires Cout % 64 == 0, Cin % 4 == 0, EXEC all-ones (full 256-thread blocks).
// ---------------------------------------------------------------------------
__global__ void __launch_bounds__(256) gemm_wmma_kernel(
    const float* __restrict__ w, const float* __restrict__ bias,
    const float* __restrict__ h, float* __restrict__ y,
    int Cout, int Cin)
{
    int lane = threadIdx.x & 31;
    int wave = threadIdx.x >> 5;
    int col0 = (blockIdx.x * 8 + wave) * 16;
    int m0   = blockIdx.y * 64;
    int b    = blockIdx.z;
    const float* hb = h + (size_t)b * Cin * NK_;
    float*       yb = y + (size_t)b * Cout * NK_;

    int half = lane >> 4;      // 0 or 1
    int mr   = lane & 15;

    v8f acc0 = {}, acc1 = {}, acc2 = {}, acc3 = {};
    for (int k0 = 0; k0 < Cin; k0 += 4) {
        v2f bv;
        bv.x = hb[(size_t)(k0 + 2 * half)     * NK_ + col0 + mr];
        bv.y = hb[(size_t)(k0 + 2 * half + 1) * NK_ + col0 + mr];

        const float* wr = w + (size_t)(m0 + mr) * Cin + k0 + 2 * half;
        v2f a0, a1, a2, a3;
        a0.x = wr[0];              a0.y = wr[1];
        a1.x = wr[16 * Cin];       a1.y = wr[16 * Cin + 1];
        a2.x = wr[32 * Cin];       a2.y = wr[32 * Cin + 1];
        a3.x = wr[48 * Cin];       a3.y = wr[48 * Cin + 1];

        acc0 = __builtin_amdgcn_wmma_f32_16x16x4_f32(false, a0, false, bv, (short)0, acc0, false, false);
        acc1 = __builtin_amdgcn_wmma_f32_16x16x4_f32(false, a1, false, bv, (short)0, acc1, false, false);
        acc2 = __builtin_amdgcn_wmma_f32_16x16x4_f32(false, a2, false, bv, (short)0, acc2, false, false);
        acc3 = __builtin_amdgcn_wmma_f32_16x16x4_f32(false, a3, false, bv, (short)0, acc3, false, false);
    }
    #pragma unroll
    for (int r = 0; r < 8; ++r) {
        int mrow = r + 8 * half;
        yb[(size_t)(m0 +      mrow) * NK_ + col0 + mr] = acc0[r] + bias[m0 +      mrow];
        yb[(size_t)(m0 + 16 + mrow) * NK_ + col0 + mr] = acc1[r] + bias[m0 + 16 + mrow];
        yb[(size_t)(m0 + 32 + mrow) * NK_ + col0 + mr] = acc2[r] + bias[m0 + 32 + mrow];
        yb[(size_t)(m0 + 48 + mrow) * NK_ + col0 + mr] = acc3[r] + bias[m0 + 48 + mrow];
    }
}

// ---------------------------------------------------------------------------
// GroupNorm stats: one block per (group, batch). Deterministic (no atomics).
// ---------------------------------------------------------------------------
__global__ void group_stats_kernel(const float* __restrict__ y,
                                   float* __restrict__ stats, int Cout)
{
    int g = blockIdx.x, b = blockIdx.y;
    int cpg = Cout / GROUPS_;
    const float* base = y + ((size_t)b * Cout + (size_t)g * cpg) * NK_;
    size_t cnt = (size_t)cpg * NK_;

    float s = 0.f, ss = 0.f;
    for (size_t i = threadIdx.x; i < cnt; i += blockDim.x) {
        float v = base[i];
        s += v; ss += v * v;
    }
    __shared__ float sh_s[256], sh_ss[256];
    sh_s[threadIdx.x] = s; sh_ss[threadIdx.x] = ss;
    __syncthreads();
    for (int st = 128; st > 0; st >>= 1) {
        if ((int)threadIdx.x < st) {
            sh_s[threadIdx.x]  += sh_s[threadIdx.x + st];
            sh_ss[threadIdx.x] += sh_ss[threadIdx.x + st];
        }
        __syncthreads();
    }
    if (threadIdx.x == 0) {
        float inv  = 1.0f / (float)cnt;
        float mean = sh_s[0] * inv;
        float var  = sh_ss[0] * inv - mean * mean;
        stats[(b * GROUPS_ + g) * 2 + 0] = mean;
        stats[(b * GROUPS_ + g) * 2 + 1] = rsqrtf(var + EPS_GN);
    }
}

// ---------------------------------------------------------------------------
// normalize + affine + LeakyReLU (elementwise)
// ---------------------------------------------------------------------------
__global__ void norm_act_kernel(const float* __restrict__ y,
                                const float* __restrict__ stats,
                                const float* __restrict__ gamma,
                                const float* __restrict__ beta,
                                float* __restrict__ out, int Cout)
{
    size_t total = (size_t)B_ * Cout * NK_;
    int cpg = Cout / GROUPS_;
    size_t c_nk = (size_t)Cout * NK_;
    for (size_t i = (size_t)blockIdx.x * blockDim.x + threadIdx.x; i < total;
         i += (size_t)gridDim.x * blockDim.x) {
        int b = (int)(i / c_nk);
        size_t rem = i - (size_t)b * c_nk;
        int c = (int)(rem / NK_);
        int g = c / cpg;
        float mean = stats[(b * GROUPS_ + g) * 2];
        float rstd = stats[(b * GROUPS_ + g) * 2 + 1];
        float v = (y[i] - mean) * rstd * gamma[c] + beta[c];
        out[i] = v >= 0.f ? v : 0.2f * v;
    }
}

// ---------------------------------------------------------------------------
// normalize + affine + LeakyReLU + max over k -> (B, C, N)
// ---------------------------------------------------------------------------
__global__ void norm_act_max_kernel(const float* __restrict__ y,
                                    const float* __restrict__ stats,
                                    const float* __restrict__ gamma,
                                    const float* __restrict__ beta,
                                    float* __restrict__ out, int Cout)
{
    int t = blockIdx.x * blockDim.x + threadIdx.x;
    if (t >= B_ * Cout * N_) return;
    int b   = t / (Cout * N_);
    int rem = t - b * (Cout * N_);
    int c   = rem / N_;
    int n   = rem - c * N_;
    int cpg = Cout / GROUPS_;
    int g   = c / cpg;
    float mean = stats[(b * GROUPS_ + g) * 2];
    float rstd = stats[(b * GROUPS_ + g) * 2 + 1];
    float ga = gamma[c], be = beta[c];
    const float* yb = y + ((size_t)b * Cout + c) * NK_ + (size_t)n * KNN_;
    float mx = -3.0e38f;
    #pragma unroll
    for (int kk = 0; kk < KNN_; ++kk) {
        float v = (yb[kk] - mean) * rstd * ga + be;
        v = v >= 0.f ? v : 0.2f * v;
        mx = fmaxf(mx, v);
    }
    out[t] = mx;
}

// ---------------------------------------------------------------------------
extern "C" void kernel_launch(void* const* d_in, const int* in_sizes, int n_in,
                              void* d_out, int out_size, void* d_ws, size_t ws_size,
                              hipStream_t stream)
{
    (void)in_sizes; (void)n_in; (void)out_size; (void)ws_size;

    const float* x    = (const float*)d_in[0];
    const float* pos  = (const float*)d_in[1];
    const float* w1a  = (const float*)d_in[2];
    const float* b1a  = (const float*)d_in[3];
    const float* g1a  = (const float*)d_in[4];
    const float* be1a = (const float*)d_in[5];
    const float* w1b  = (const float*)d_in[6];
    const float* b1b  = (const float*)d_in[7];
    const float* g1b  = (const float*)d_in[8];
    const float* be1b = (const float*)d_in[9];
    const float* w2a  = (const float*)d_in[10];
    const float* b2a  = (const float*)d_in[11];
    const float* g2a  = (const float*)d_in[12];
    const float* be2a = (const float*)d_in[13];
    const float* w2b  = (const float*)d_in[14];
    const float* b2b  = (const float*)d_in[15];
    const float* g2b  = (const float*)d_in[16];
    const float* be2b = (const float*)d_in[17];

    char* ws = (char*)d_ws;
    const size_t MB = 1u << 20;
    int*   idx   = (int*)  (ws + 0);                // 0.66 MB
    float* stats = (float*)(ws + 1 * MB);           // tiny
    float* bufA  = (float*)(ws + 2 * MB);           // up to 86.5 MB (h / activations)
    float* bufB  = (float*)(ws + 90 * MB);          // up to 84 MB (GEMM out)
    float* x1    = (float*)(ws + 174 * MB);         // 2 MB

    // kNN once — pos is identical for both edge_conv layers.
    knn_kernel<<<(B_ * N_ + 127) / 128, 128, 0, stream>>>(pos, idx);

    dim3 blk(256);
    dim3 gGF((B_ * N_ * KNN_ + 255) / 256);
    dim3 gStats(GROUPS_, B_);

    // ---- edge_conv 1 ----
    graph_feature_kernel<<<gGF, blk, 0, stream>>>(x, pos, idx, bufA, CIN_);
    dim3 gG1(NK_ / 16 / 8, 64 / 64, B_);                       // (640, 1, 2)
    gemm_wmma_kernel<<<gG1, blk, 0, stream>>>(w1a, b1a, bufA, bufB, 64, 36);
    group_stats_kernel<<<gStats, blk, 0, stream>>>(bufB, stats, 64);
    norm_act_kernel<<<2048, blk, 0, stream>>>(bufB, stats, g1a, be1a, bufA, 64);
    gemm_wmma_kernel<<<gG1, blk, 0, stream>>>(w1b, b1b, bufA, bufB, 64, 64);
    group_stats_kernel<<<gStats, blk, 0, stream>>>(bufB, stats, 64);
    norm_act_max_kernel<<<(B_ * 64 * N_ + 255) / 256, blk, 0, stream>>>(
        bufB, stats, g1b, be1b, x1, 64);

    // ---- edge_conv 2 (same idx) ----
    graph_feature_kernel<<<gGF, blk, 0, stream>>>(x1, pos, idx, bufA, 64);
    dim3 gG2(NK_ / 16 / 8, 128 / 64, B_);                      // (640, 2, 2)
    gemm_wmma_kernel<<<gG2, blk, 0, stream>>>(w2a, b2a, bufA, bufB, 128, 132);
    group_stats_kernel<<<gStats, blk, 0, stream>>>(bufB, stats, 128);
    norm_act_kernel<<<4096, blk, 0, stream>>>(bufB, stats, g2a, be2a, bufA, 128);
    gemm_wmma_kernel<<<gG2, blk, 0, stream>>>(w2b, b2b, bufA, bufB, 128, 128);
    group_stats_kernel<<<gStats, blk, 0, stream>>>(bufB, stats, 128);
    norm_act_max_kernel<<<(B_ * 128 * N_ + 255) / 256, blk, 0, stream>>>(
        bufB, stats, g2b, be2b, (float*)d_out, 128);
}